// EmbeddingAlignment_72310069395865
// MI455X (gfx1250) — compile-verified
//
#include <hip/hip_runtime.h>
#include <math.h>

typedef __attribute__((ext_vector_type(2))) float v2f;
typedef __attribute__((ext_vector_type(8))) float v8f;
typedef __attribute__((__vector_size__(4 * sizeof(int)))) int v4i;

// D = A(16x4) * B(4x16) + C, fp32 WMMA (CDNA5)
#define WMMA_F32(a, b, c) \
  __builtin_amdgcn_wmma_f32_16x16x4_f32(false, (a), false, (b), (short)0, (c), false, false)

#if defined(__has_builtin)
#if __has_builtin(__builtin_amdgcn_global_load_async_to_lds_b128)
#define ASYNC_LDS 1
#endif
#if __has_builtin(__builtin_amdgcn_s_wait_asynccnt)
#define HAVE_WAIT_ASYNC_BUILTIN 1
#endif
#endif

// Async global->LDS b128 copy (CDNA5 ASYNCcnt-tracked path, bypasses VGPRs),
// with a VGPR-roundtrip fallback if the builtin is unavailable.
__device__ __forceinline__ void copy_b128_to_lds(const float* gsrc, float* ldst) {
#ifdef ASYNC_LDS
  __builtin_amdgcn_global_load_async_to_lds_b128(
      (__attribute__((address_space(1))) v4i*)gsrc,
      (__attribute__((address_space(3))) v4i*)ldst, 0, 0);
#else
  *(float4*)ldst = *(const float4*)gsrc;
#endif
}

__device__ __forceinline__ void copy_wait() {
#ifdef ASYNC_LDS
#ifdef HAVE_WAIT_ASYNC_BUILTIN
  __builtin_amdgcn_s_wait_asynccnt(0);
#else
  asm volatile("s_wait_asynccnt 0x0" ::: "memory");
#endif
#endif
}

constexpr int kB = 32;       // batch
constexpr int kC = 256;      // channels (K dim)
constexpr int kM = 1024;     // keys   (H*W)
constexpr int kN = 1024;     // queries
constexpr float kScale = 1.0f / 16.0f;  // 1/sqrt(C)

// ---------------------------------------------------------------------------
// Pass 1: per (batch, 32-column strip), compute attn = q @ kv^T / 16 over all
// 1024 rows and reduce column logsumexp -> colL[b][m].
// kv[m][c] = src[b][c][m], so B-fragment (K x N layout, lane = column m) loads
// straight from src with perfect coalescing. q is staged through LDS.
// ---------------------------------------------------------------------------
__global__ __launch_bounds__(256)
void colstats_kernel(const float* __restrict__ src, const float* __restrict__ q,
                     float* __restrict__ colL) {
  const int b     = blockIdx.y;
  const int mbase = blockIdx.x * 32;
  const float* __restrict__ srcb = src + (size_t)b * kC * kM;

  __shared__ float qs[128 * 68];   // [128 n-rows][64 k + 4 pad] : 34.8 KB
  __shared__ float smax[8 * 32];
  __shared__ float ssum[8 * 32];

  const int tid  = threadIdx.x;
  const int wid  = tid >> 5;
  const int lane = tid & 31;
  const int half = lane >> 4;
  const int l16  = lane & 15;

  // running column logsumexp state, col = mbase + mt*16 + l16 (dup'd in halves)
  float cmax0 = -__builtin_inff(), cmax1 = -__builtin_inff();
  float csum0 = 0.f, csum1 = 0.f;

  for (int nchunk = 0; nchunk < 8; ++nchunk) {
    const int nrow0 = nchunk * 128;
    v8f d0 = {}; v8f d1 = {};
    for (int ks = 0; ks < 4; ++ks) {
      const int k0 = ks * 64;
      __syncthreads();  // previous stage fully consumed
      // cooperative stage q[nrow0..+128][k0..+64] -> LDS (async b128, coalesced)
      {
        const int c4 = (tid & 15) * 4;   // 0..60
        const int r0 = tid >> 4;         // 0..15
#pragma unroll
        for (int s = 0; s < 8; ++s) {
          const int rr = r0 + s * 16;
          copy_b128_to_lds(q + (size_t)(nrow0 + rr) * kC + k0 + c4,
                           qs + rr * 68 + c4);
        }
        copy_wait();
      }
      __syncthreads();
      const int arow = wid * 16 + l16;
#pragma unroll 4
      for (int k = 0; k < 64; k += 4) {
        // A-frag: q row (n), v[0]=K, v[1]=K+1, halves offset by 2
        v2f a = *(const v2f*)(qs + arow * 68 + k + half * 2);
        // B-frag: B[k][m] = src[b][k][m]; lanes sweep m -> coalesced
        const float* bp = srcb + (size_t)(k0 + k + half * 2) * kM + mbase + l16;
        v2f b0; b0.x = bp[0];  b0.y = bp[kM];
        v2f b1; b1.x = bp[16]; b1.y = bp[kM + 16];
        d0 = WMMA_F32(a, b0, d0);
        d1 = WMMA_F32(a, b1, d1);
      }
    }
    // per-16-row-chunk column stats: lane holds rows r+8*half of its column
    float tmax0 = -__builtin_inff(), tmax1 = -__builtin_inff();
#pragma unroll
    for (int r = 0; r < 8; ++r) {
      tmax0 = fmaxf(tmax0, d0[r] * kScale);
      tmax1 = fmaxf(tmax1, d1[r] * kScale);
    }
    float tsum0 = 0.f, tsum1 = 0.f;
#pragma unroll
    for (int r = 0; r < 8; ++r) {
      tsum0 += expf(d0[r] * kScale - tmax0);
      tsum1 += expf(d1[r] * kScale - tmax1);
    }
    // merge the two 8-row halves (lane ^ 16 holds the other 8 rows)
    {
      float om = __shfl_xor(tmax0, 16, 32), os = __shfl_xor(tsum0, 16, 32);
      float nm = fmaxf(tmax0, om);
      tsum0 = tsum0 * expf(tmax0 - nm) + os * expf(om - nm);
      tmax0 = nm;
      om = __shfl_xor(tmax1, 16, 32); os = __shfl_xor(tsum1, 16, 32);
      nm = fmaxf(tmax1, om);
      tsum1 = tsum1 * expf(tmax1 - nm) + os * expf(om - nm);
      tmax1 = nm;
    }
    // online merge into running state
    {
      float nm = fmaxf(cmax0, tmax0);
      csum0 = csum0 * expf(cmax0 - nm) + tsum0 * expf(tmax0 - nm);
      cmax0 = nm;
      nm = fmaxf(cmax1, tmax1);
      csum1 = csum1 * expf(cmax1 - nm) + tsum1 * expf(tmax1 - nm);
      cmax1 = nm;
    }
  }
  // cross-wave reduce through LDS
  if (half == 0) {
    smax[wid * 32 + l16]      = cmax0;  ssum[wid * 32 + l16]      = csum0;
    smax[wid * 32 + 16 + l16] = cmax1;  ssum[wid * 32 + 16 + l16] = csum1;
  }
  __syncthreads();
  if (tid < 32) {
    float rm = smax[tid], rs = ssum[tid];
    for (int w = 1; w < 8; ++w) {
      const float om = smax[w * 32 + tid], os = ssum[w * 32 + tid];
      const float nm = fmaxf(rm, om);
      rs = rs * expf(rm - nm) + os * expf(om - nm);
      rm = nm;
    }
    colL[b * kM + mbase + tid] = rm + logf(rs);  // column logsumexp
  }
}

// ---------------------------------------------------------------------------
// Pass 2: per (batch, 32-row strip), recompute attn rows, form
// score = 2*attn - colL[m], running argmax over m per row, then gather
// out[b][c][n] = src[b][c][m*(n)].
// ---------------------------------------------------------------------------
__global__ __launch_bounds__(256)
void argmax_gather_kernel(const float* __restrict__ src,
                          const float* __restrict__ q,
                          const float* __restrict__ colL,
                          float* __restrict__ out) {
  const int b     = blockIdx.y;
  const int nbase = blockIdx.x * 32;
  const float* __restrict__ srcb = src + (size_t)b * kC * kM;

  __shared__ float qs[32 * 260];  // [32 n-rows][256 k + 4 pad] : 33.3 KB
  __shared__ float sS[8 * 32];
  __shared__ int   sI[8 * 32];
  __shared__ int   bestm[32];

  const int tid  = threadIdx.x;
  const int wid  = tid >> 5;
  const int lane = tid & 31;
  const int half = lane >> 4;
  const int l16  = lane & 15;

  // stage q[nbase..+32][0..256] -> LDS (async b128, coalesced)
  {
    const int c4 = (tid & 63) * 4;  // 0..252
    const int r0 = tid >> 6;        // 0..3
#pragma unroll
    for (int s = 0; s < 8; ++s) {
      const int rr = r0 + s * 4;
      copy_b128_to_lds(q + (size_t)(nbase + rr) * kC + c4, qs + rr * 260 + c4);
    }
    copy_wait();
  }
  __syncthreads();

  // running per-row argmax; slot nt*8+r <-> row nt*16 + 8*half + r
  float bs[16]; int bi[16];
#pragma unroll
  for (int i = 0; i < 16; ++i) { bs[i] = -__builtin_inff(); bi[i] = 0; }

  const int mstart = wid * 128;  // waves split the m dimension
  for (int mt = 0; mt < 8; ++mt) {
    const int mbase = mstart + mt * 16;
    v8f d0 = {}; v8f d1 = {};
#pragma unroll 4
    for (int k = 0; k < 256; k += 4) {
      const int kk = k + half * 2;
      v2f a0 = *(const v2f*)(qs + (0  + l16) * 260 + kk);
      v2f a1 = *(const v2f*)(qs + (16 + l16) * 260 + kk);
      const float* bp = srcb + (size_t)kk * kM + mbase + l16;
      v2f bb; bb.x = bp[0]; bb.y = bp[kM];
      d0 = WMMA_F32(a0, bb, d0);  // rows nbase..+15
      d1 = WMMA_F32(a1, bb, d1);  // rows nbase+16..+31
    }
    const float cl = colL[b * kM + mbase + l16];
#pragma unroll
    for (int nt = 0; nt < 2; ++nt) {
#pragma unroll
      for (int r = 0; r < 8; ++r) {
        float s = (nt ? d1[r] : d0[r]) * (2.0f * kScale) - cl;
        int idx = mbase + l16;
        // argmax across the 16 m-lanes of this half
#pragma unroll
        for (int off = 1; off < 16; off <<= 1) {
          const float os = __shfl_xor(s, off, 32);
          const int   oi = __shfl_xor(idx, off, 32);
          if (os > s || (os == s && oi < idx)) { s = os; idx = oi; }
        }
        const int slot = nt * 8 + r;
        if (s > bs[slot] || (s == bs[slot] && idx < bi[slot])) {
          bs[slot] = s; bi[slot] = idx;
        }
      }
    }
  }
  // cross-wave argmax through LDS
  if (l16 == 0) {
#pragma unroll
    for (int nt = 0; nt < 2; ++nt)
#pragma unroll
      for (int r = 0; r < 8; ++r) {
        const int row = nt * 16 + 8 * half + r;
        sS[wid * 32 + row] = bs[nt * 8 + r];
        sI[wid * 32 + row] = bi[nt * 8 + r];
      }
  }
  __syncthreads();
  if (tid < 32) {
    float rS = sS[tid]; int rI = sI[tid];
    for (int w = 1; w < 8; ++w) {
      const float oS = sS[w * 32 + tid]; const int oI = sI[w * 32 + tid];
      if (oS > rS || (oS == rS && oI < rI)) { rS = oS; rI = oI; }
    }
    bestm[tid] = rI;
  }
  __syncthreads();
  // gather: out[b][c][nbase+nloc] = src[b][c][m*]; lanes sweep n -> coalesced
  {
    const int nloc = tid & 31;
    const int c0   = tid >> 5;  // 0..7
    const int m    = bestm[nloc];
    float* __restrict__ outb = out + (size_t)b * kC * kN + nbase + nloc;
#pragma unroll 4
    for (int c = c0; c < kC; c += 8) {
      outb[(size_t)c * kN] = srcb[(size_t)c * kM + m];
    }
  }
}

extern "C" void kernel_launch(void* const* d_in, const int* in_sizes, int n_in,
                              void* d_out, int out_size, void* d_ws,
                              size_t ws_size, hipStream_t stream) {
  (void)in_sizes; (void)n_in; (void)out_size; (void)ws_size;
  const float* src = (const float*)d_in[0];   // [B, C, H, W] = [32,256,32,32]
  const float* qw  = (const float*)d_in[1];   // [N, C] = [1024, 256]
  float* out  = (float*)d_out;                // [B, C, H, W]
  float* colL = (float*)d_ws;                 // B*M floats = 128 KB scratch

  const dim3 blk(256);
  colstats_kernel<<<dim3(kM / 32, kB), blk, 0, stream>>>(src, qw, colL);
  argmax_gather_kernel<<<dim3(kN / 32, kB), blk, 0, stream>>>(src, qw, colL, out);
}